// SparseLayer_69827578298456
// MI455X (gfx1250) — compile-verified
//
#include <hip/hip_runtime.h>

// SparseLayer on MI455X (gfx1250, wave32).
// out[b, o] = sum_{k<8} x[b, k*512 + (o%512)] * W[o, k*512 + (o%512)] + bias[o]
// Memory-bound: 256 MB unavoidable traffic -> ~11 us floor at 23.3 TB/s.
// Per (16-batch-rows, c) tile: (16x8) = (16x8 A) x (8x8 B) done as two chained
// V_WMMA_F32_16X16X4_F32 with N zero-padded to 16. LDS staging on both sides
// keeps all global traffic as coalesced 128-bit non-temporal transactions.

typedef __attribute__((ext_vector_type(2))) float v2f;
typedef __attribute__((ext_vector_type(4))) float v4f;
typedef __attribute__((ext_vector_type(8))) float v8f;

#define SIZE_IN   4096
#define SIZE_OUT  4096
#define GROUPS    512      // inp_el_counts = SIZE_IN / STEPS
#define STEPS     8
#define BATCH     8192

#define CBLK      32       // c-values handled per workgroup
#define BTILE     16       // batch rows per tile (WMMA M)
#define NTHREADS  256      // 8 wave32s

// 260 floats/row: 16B-aligned row base (b128 LDS ops) and 260%64=4 spreads
// A-read banks (lanes 0-15 hit banks 4m%64, all distinct).
#define XS_STRIDE 260
#define OS_SEG    40                   // 40%64 spreads D-store banks; 160B row => 16B aligned
#define OS_MSTRIDE (8 * OS_SEG)        // 320

__global__ __launch_bounds__(NTHREADS)
void sparse_wmma_kernel(const float* __restrict__ x,
                        const float* __restrict__ W,
                        const float* __restrict__ bias,
                        const float* __restrict__ mask,
                        float* __restrict__ out,
                        int tiles_per_wg)
{
    __shared__ float xs[BTILE * XS_STRIDE];     // x tile:   [m][k*32 + cc]
    __shared__ float ws[CBLK * 8 * 16];         // B image:  [cidx][k][j(0..15, >=8 zero)]
    __shared__ float os[BTILE * OS_MSTRIDE];    // out tile: [m][j][cc]

    const int tid    = threadIdx.x;
    const int c_base = blockIdx.x * CBLK;

    // ---- gather masked weights into zero-padded LDS B-matrix image (once per WG) ----
    // Scattered 4B reads; W/mask lines stay L2-resident across workgroups (default TH).
    #pragma unroll
    for (int i = 0; i < (CBLK * 8 * 16) / NTHREADS; ++i) {   // 16 iters
        int flat = i * NTHREADS + tid;
        int cidx = flat >> 7;          // /128
        int k    = (flat >> 4) & 7;
        int j    = flat & 15;
        int c    = c_base + cidx;
        float v  = 0.0f;
        if (j < 8) {
            size_t o    = (size_t)(c + j * GROUPS);
            size_t icol = (size_t)(c + k * GROUPS);
            v = W[o * SIZE_IN + icol] * mask[o * SIZE_IN + icol];
        }
        ws[flat] = v;
    }
    __syncthreads();

    const int wave    = tid >> 5;        // 0..7
    const int lane    = tid & 31;
    const int halfsel = lane >> 4;       // 0: lanes 0-15, 1: lanes 16-31
    const int n       = lane & 15;       // WMMA N index (also A's M index)
    const int kb      = halfsel * 2;     // A/B K sub-offset per ISA f32 layout

    for (int t = 0; t < tiles_per_wg; ++t) {
        const int btile = blockIdx.y + t * gridDim.y;
        const int b0    = btile * BTILE;

        // ---- stage x tile into LDS: 128-bit coalesced non-temporal loads ----
        // 16 rows x 64 float4 = 1024 float4 -> 4 iterations of 256 threads.
        #pragma unroll
        for (int i = 0; i < (BTILE * (8 * CBLK) / 4) / NTHREADS; ++i) {
            int flat4 = i * NTHREADS + tid;
            int m     = flat4 >> 6;        // 64 float4 per row
            int q     = flat4 & 63;        // float4 index in row; col = q*4
            int k     = q >> 3;            // segment (col/32)
            int cc    = (q & 7) * 4;       // offset in segment, 16B aligned
            const v4f* src = reinterpret_cast<const v4f*>(
                &x[(size_t)(b0 + m) * SIZE_IN + (size_t)(k * GROUPS + c_base + cc)]);
            v4f v = __builtin_nontemporal_load(src);
            *reinterpret_cast<v4f*>(&xs[m * XS_STRIDE + q * 4]) = v;
        }
        __syncthreads();

        // ---- compute: each wave covers CBLK/8 = 4 c-values ----
        for (int i = 0; i < CBLK / 8; ++i) {
            const int cidx = wave * (CBLK / 8) + i;

            // A (16x4 f32): lane L<16 holds {K=0,K=1}, L>=16 holds {K=2,K=3}; M = lane%16
            const float* xrow = &xs[n * XS_STRIDE + cidx];
            v2f a1, a2;
            a1.x = xrow[(kb + 0) * CBLK];
            a1.y = xrow[(kb + 1) * CBLK];
            a2.x = xrow[(kb + 4) * CBLK];
            a2.y = xrow[(kb + 5) * CBLK];

            // B (4x16 f32): VGPR0 = rows K=0 (lanes 0-15) / K=2 (lanes 16-31); VGPR1 = K=1/K=3
            const float* wrow = &ws[cidx * 128 + n];
            v2f b1, b2;
            b1.x = wrow[(kb + 0) * 16];
            b1.y = wrow[(kb + 1) * 16];
            b2.x = wrow[(kb + 4) * 16];
            b2.y = wrow[(kb + 5) * 16];

            v8f acc = {};
            acc = __builtin_amdgcn_wmma_f32_16x16x4_f32(
                      false, a1, false, b1, (short)0, acc, false, false);
            acc = __builtin_amdgcn_wmma_f32_16x16x4_f32(
                      false, a2, false, b2, (short)0, acc, false, false);

            // D: VGPR r -> M = r (lanes 0-15) / r+8 (lanes 16-31), N = lane%16
            if (n < 8) {
                #pragma unroll
                for (int r = 0; r < 8; ++r) {
                    int m = r + halfsel * 8;
                    os[m * OS_MSTRIDE + n * OS_SEG + cidx] = acc[r];
                }
            }
        }
        __syncthreads();

        // ---- write out tile (+bias): 128-bit coalesced non-temporal stores ----
        #pragma unroll
        for (int i = 0; i < (BTILE * (8 * CBLK) / 4) / NTHREADS; ++i) {
            int flat4 = i * NTHREADS + tid;
            int m     = flat4 >> 6;
            int q     = flat4 & 63;
            int j     = q >> 3;
            int cc    = (q & 7) * 4;
            int o     = j * GROUPS + c_base + cc;

            v4f v  = *reinterpret_cast<const v4f*>(
                &os[m * OS_MSTRIDE + j * OS_SEG + cc]);
            v4f bb = *reinterpret_cast<const v4f*>(&bias[o]);
            v += bb;

            v4f* dst = reinterpret_cast<v4f*>(
                &out[(size_t)(b0 + m) * SIZE_OUT + o]);
            __builtin_nontemporal_store(v, dst);
        }
        __syncthreads();
    }
}

extern "C" void kernel_launch(void* const* d_in, const int* in_sizes, int n_in,
                              void* d_out, int out_size, void* d_ws, size_t ws_size,
                              hipStream_t stream) {
    const float* x    = (const float*)d_in[0];   // [BATCH, SIZE_IN]
    const float* W    = (const float*)d_in[1];   // [SIZE_OUT, SIZE_IN]
    const float* bias = (const float*)d_in[2];   // [SIZE_OUT]
    const float* mask = (const float*)d_in[3];   // [SIZE_OUT, SIZE_IN]
    float* out        = (float*)d_out;           // [BATCH, SIZE_OUT]

    const int c_blocks     = GROUPS / CBLK;              // 16
    const int batch_tiles  = BATCH / BTILE;              // 512
    const int grid_y       = 128;
    const int tiles_per_wg = batch_tiles / grid_y;       // 4

    dim3 grid(c_blocks, grid_y);
    dim3 block(NTHREADS);
    sparse_wmma_kernel<<<grid, block, 0, stream>>>(x, W, bias, mask, out, tiles_per_wg);
}